// FullSpikeFunction_50826642981383
// MI455X (gfx1250) — compile-verified
//
#include <hip/hip_runtime.h>
#include <stdint.h>

// Adaptive-LIF spiking forward scan, MI455X (gfx1250, wave32).
//
// Roofline: 512 MiB touched once @ 23.3 TB/s -> ~22 us floor; FLOPs trivial.
// Elementwise recurrence over T (no GEMM -> WMMA inapplicable). This version
// streams x through LDS with CDNA5 GLOBAL_LOAD_ASYNC_TO_LDS_B128 (ASYNCcnt),
// ring-buffered 3 chunks (~96 rows) ahead of the dependent scan chain, so
// in-flight traffic is committed DMA rather than best-effort prefetch, and no
// VGPRs are spent on x double-buffering.

namespace {

constexpr float AL        = 0.9f;                  // RESET_ALPHA
constexpr float ONE_M_AL  = (float)(1.0 - 0.9);    // match JAX double->f32 folding
constexpr float BET       = 0.95f;                 // THR_ALPHA
constexpr float ONE_M_BET = (float)(1.0 - 0.95);
constexpr float REFRAC    = 5.0f;

constexpr int LPW   = 32;            // lanes per wave (wave32)
constexpr int WPB   = 2;             // waves per block
constexpr int BLOCK = WPB * LPW;     // 64 threads
constexpr int U     = 32;            // time rows per chunk
constexpr int D     = 4;             // ring slots (chunks in flight), pow2
constexpr int ILP   = 2;             // independent batches per lane
constexpr int AS_PER_CHUNK = (U / 4) * ILP;           // 16 b128 issues per chunk
constexpr int WAIT_STEADY  = (D - 1) * AS_PER_CHUNK;  // 48 <= 63 counter cap

typedef int v4i __attribute__((ext_vector_type(4)));

struct SeqState {
  float u, b, q, rz;  // rz carries last_z * reset_gamma[n]
  float rg, cc;       // reset_gamma[n], (1-beta)*thr_gamma[n]
};

// One scan step. Reference semantics:
//   u = a*u + (1-a)*x - last_z*rg ; z = (u-b > 0) masked by (q == 0)
//   q = max(q-1,0) + 5*z          ; b = B*b + (1-B)*z*tg
__device__ __forceinline__ float alif_step(SeqState& s, float xv) {
  s.u = fmaf(AL, s.u, fmaf(ONE_M_AL, xv, -s.rz));
  float v = s.u - s.b;
  bool fire = (v > 0.0f) && (s.q < 0.5f);  // q in {0..5}, exact in f32
  float z  = fire ? 1.0f : 0.0f;
  float qd = fmaxf(s.q - 1.0f, 0.0f);
  s.q  = fire ? REFRAC : qd;
  s.b  = fmaf(BET, s.b, fire ? s.cc : 0.0f);
  s.rz = fire ? s.rg : 0.0f;
  return z;
}

// ---- CDNA5 async global->LDS (ASYNCcnt-tracked DMA) ----
// Probe-learned signature: (v4i AS1*, v4i AS3*, imm offset, imm cpol).
__device__ __forceinline__ void async_load_b128(const float* gsrc, float* lds) {
#if __has_builtin(__builtin_amdgcn_global_load_async_to_lds_b128)
  __builtin_amdgcn_global_load_async_to_lds_b128(
      (__attribute__((address_space(1))) v4i*)gsrc,
      (__attribute__((address_space(3))) v4i*)lds, 0, 0);
#else
  unsigned laddr = (unsigned)(uintptr_t)lds;  // LDS byte offset (aperture low bits)
  asm volatile("global_load_async_to_lds_b128 %0, %1, off"
               :: "v"(laddr), "v"(gsrc)
               : "memory");
#endif
}

template <int CNT>
__device__ __forceinline__ void wait_asynccnt() {
#if __has_builtin(__builtin_amdgcn_s_wait_asynccnt)
  __builtin_amdgcn_s_wait_asynccnt(CNT);
#else
  asm volatile("s_wait_asynccnt %0" :: "i"(CNT) : "memory");
#endif
  asm volatile("" ::: "memory");  // keep LDS reads below the wait
}

__global__ __launch_bounds__(BLOCK) void alif_scan_async_kernel(
    const float* __restrict__ x,
    const float* __restrict__ reset_gamma,
    const float* __restrict__ thr_gamma,
    const float* __restrict__ u0,
    const float* __restrict__ b0,
    float* __restrict__ zout,
    int B, int T, int N)
{
  // Wave-private ring tiles: [wave][slot][k][row*32 + col]; 64 KB/block.
  __shared__ __align__(16) float tiles[WPB][D][ILP][U * LPW];

  const int half = B / ILP;
  const int seqs = half * N;
  const int g = blockIdx.x * BLOCK + threadIdx.x;
  if (g >= seqs) return;

  const int lane = threadIdx.x & (LPW - 1);
  const int w    = threadIdx.x >> 5;
  const int n    = g % N;          // lanes hold consecutive n (coalesced rows)
  const int b    = g / N;          // uniform across the wave (32 | N)
  const int n0   = n - lane;       // wave-base column
  const long row = (long)N;        // element stride between consecutive t

  // Staging geometry: one b128 moves 4 rows x 32 cols (512B). Lane -> 16B unit:
  const int  sub  = lane >> 3;                        // row within 4-row group
  const int  quad = lane & 7;                         // 16B unit within row
  const long lane_goff = (long)sub * row + quad * 4;  // global elem offset
  const int  lane_loff = sub * LPW + quad * 4;        // LDS elem offset

  SeqState st[ILP];
  const float* gsrc[ILP];  // next chunk to ISSUE (row base, lane-adjusted)
  float*       zp[ILP];    // next chunk to consume (store pointer)

#pragma unroll
  for (int k = 0; k < ILP; ++k) {
    const int bk = b + k * half;
    st[k].u  = u0[n];
    st[k].b  = b0[n];
    st[k].q  = 0.0f;
    st[k].rz = 0.0f;
    st[k].rg = reset_gamma[n];
    st[k].cc = ONE_M_BET * thr_gamma[n];
    const size_t base = (size_t)bk * (size_t)T * (size_t)N;
    gsrc[k] = x + base + (size_t)n0 + (size_t)lane_goff;
    zp[k]   = zout + base + (size_t)n;
  }

  const int nch = T / U;

  auto issue_chunk = [&](int slot) {
#pragma unroll
    for (int k = 0; k < ILP; ++k) {
#pragma unroll
      for (int s = 0; s < U / 4; ++s)
        async_load_b128(gsrc[k] + (long)s * 4 * row,
                        &tiles[w][slot][k][s * 4 * LPW + lane_loff]);
      gsrc[k] += (long)U * row;
    }
  };

  auto consume_chunk = [&](int ch) {
    const int slot = ch & (D - 1);
#pragma unroll
    for (int j = 0; j < U; ++j) {
#pragma unroll
      for (int k = 0; k < ILP; ++k) {            // interleave 2 chains (dual-issue)
        float xv = tiles[w][slot][k][j * LPW + lane];
        float z  = alif_step(st[k], xv);
        __builtin_nontemporal_store(z, zp[k] + (long)j * row);
      }
    }
#pragma unroll
    for (int k = 0; k < ILP; ++k) zp[k] += (long)U * row;
  };

  // Prologue: commit D-1 chunks of DMA before any compute.
  const int pre = (nch < D - 1) ? nch : (D - 1);
  for (int i = 0; i < pre; ++i) issue_chunk(i);

  // Steady state: issue chunk ch+D-1, wait until only the 3 newest chunks'
  // DMAs are outstanding (in-order completion => chunk ch has landed).
  int ch = 0;
  for (; ch + (D - 1) < nch; ++ch) {
    issue_chunk((ch + D - 1) & (D - 1));
    wait_asynccnt<WAIT_STEADY>();
    consume_chunk(ch);
  }
  // Epilogue: drain remaining slots (full drain; negligible at the tail).
  for (; ch < nch; ++ch) {
    wait_asynccnt<0>();
    consume_chunk(ch);
  }

  // Tail rows (T % U) — not hit for T=4096, kept for generality.
  for (int t = nch * U; t < T; ++t) {
#pragma unroll
    for (int k = 0; k < ILP; ++k) {
      const int bk = b + k * half;
      const float* xt = x + ((size_t)bk * (size_t)T + (size_t)t) * (size_t)N + n;
      float z = alif_step(st[k], __builtin_nontemporal_load(xt));
      __builtin_nontemporal_store(z, zp[k]);
      zp[k] += row;
    }
  }
}

}  // namespace

extern "C" void kernel_launch(void* const* d_in, const int* in_sizes, int n_in,
                              void* d_out, int out_size, void* d_ws, size_t ws_size,
                              hipStream_t stream) {
  const float* x  = (const float*)d_in[0];
  const float* rg = (const float*)d_in[1];
  const float* tg = (const float*)d_in[2];
  const float* u0 = (const float*)d_in[3];
  const float* b0 = (const float*)d_in[4];
  float* z = (float*)d_out;

  const int N = in_sizes[1];                                     // 1024
  const int B = 16;                                              // per setup_inputs
  const int T = (int)((long)in_sizes[0] / ((long)B * (long)N));  // 4096

  const int seqs = (B / ILP) * N;                 // 8192 threads = 256 waves
  const int grid = (seqs + BLOCK - 1) / BLOCK;    // 128 blocks of 64

  alif_scan_async_kernel<<<grid, BLOCK, 0, stream>>>(x, rg, tg, u0, b0, z, B, T, N);
}